// Model_67817533604348
// MI455X (gfx1250) — compile-verified
//
#include <hip/hip_runtime.h>

#define N_SAMPLES 32768
#define N_ITERS   16
#define KERNEL_N  256
#define N_FRAMES  512
#define N_ATOMS   256
#define STEP      (N_SAMPLES / N_FRAMES)   // 64

typedef __attribute__((ext_vector_type(2))) float v2f;
typedef __attribute__((ext_vector_type(8))) float v8f;

// Single fused workgroup:
//   stage 1: per-row argmax of atom_selection [16x256] and sched_params [16x512]
//   stage 2: atoms[16x256] = onehot(sel) @ dict  via V_WMMA_F32_16X16X4_F32 (exact in f32)
//   stage 3: scatter amp_i * atom_i at offset 64*argmax(sched_i) into 128KB LDS accumulator
//   stage 4: global max-abs + normalize, store 32768 floats
__global__ __launch_bounds__(1024) void sparse_synth_kernel(
    const float* __restrict__ atom_selection,  // [16,256]
    const float* __restrict__ amps,            // [16]
    const float* __restrict__ sched,           // [16,512]
    const float* __restrict__ dict,            // [256,256] row-major
    float* __restrict__ out)                   // [32768]
{
    __shared__ float s_acc[N_SAMPLES];              // 128 KB signal accumulator (fits in 320KB WGP LDS)
    __shared__ float s_atoms[N_ITERS][KERNEL_N];    // 16 KB
    __shared__ int   s_amax_atom[N_ITERS];
    __shared__ int   s_pos[N_ITERS];
    __shared__ float s_amp[N_ITERS];
    __shared__ float s_red[32];
    __shared__ float s_mx;

    const int tid  = threadIdx.x;
    const int lane = tid & 31;
    const int wave = tid >> 5;

    // ---- zero accumulator ----
    for (int i = tid; i < N_SAMPLES; i += 1024) s_acc[i] = 0.0f;

    // ---- stage 1: wave-parallel row argmax (first-occurrence tiebreak) ----
    if (wave < N_ITERS) {
        const float* row = atom_selection + wave * N_ATOMS;
        float bv = -3.402823466e+38f; int bi = 0;
        for (int j = lane; j < N_ATOMS; j += 32) {
            float v = row[j];
            if (v > bv) { bv = v; bi = j; }
        }
        #pragma unroll
        for (int off = 16; off > 0; off >>= 1) {
            float ov = __shfl_xor(bv, off, 32);
            int   oi = __shfl_xor(bi, off, 32);
            if (ov > bv || (ov == bv && oi < bi)) { bv = ov; bi = oi; }
        }
        if (lane == 0) {
            s_amax_atom[wave] = bi;
            s_amp[wave]       = amps[wave];
        }
    } else {
        const int r = wave - N_ITERS;
        const float* row = sched + r * N_FRAMES;
        float bv = -3.402823466e+38f; int bi = 0;
        for (int j = lane; j < N_FRAMES; j += 32) {
            float v = row[j];
            if (v > bv) { bv = v; bi = j; }
        }
        #pragma unroll
        for (int off = 16; off > 0; off >>= 1) {
            float ov = __shfl_xor(bv, off, 32);
            int   oi = __shfl_xor(bi, off, 32);
            if (ov > bv || (ov == bv && oi < bi)) { bv = ov; bi = oi; }
        }
        if (lane == 0) s_pos[r] = bi * STEP;
    }
    __syncthreads();

    // ---- stage 2: atoms = onehot @ dict via f32 WMMA (16 waves, one N-tile each) ----
    // A 16x4 layout: lanes 0-15 -> M=lane, K0 in v0 / K1 in v1; lanes 16-31 -> M=lane-16, K2/K3.
    // B 4x16 layout: lanes 0-15 -> N=lane, rows K0(v0)/K1(v1); lanes 16-31 -> N=lane-16, K2/K3.
    // C/D: VGPR r -> M=r (lanes 0-15) and M=r+8 (lanes 16-31), N = lane%16.
    if (wave < 16) {
        const int ncol  = wave * 16;       // N-tile base column
        const int mrow  = lane & 15;       // A-row (M) for this lane == B-col (N) for this lane
        const int khalf = (lane >> 4) * 2; // K sub-offset: {0,1} or {2,3}
        const int sel_k = s_amax_atom[mrow];

        v8f c = {};
        for (int kb = 0; kb < N_ATOMS; kb += 4) {
            const int k0 = kb + khalf;
            const int k1 = k0 + 1;
            v2f a, b;
            a.x = (k0 == sel_k) ? 1.0f : 0.0f;   // synthesize one-hot A in-register
            a.y = (k1 == sel_k) ? 1.0f : 0.0f;
            b.x = dict[k0 * KERNEL_N + ncol + mrow];
            b.y = dict[k1 * KERNEL_N + ncol + mrow];
            c = __builtin_amdgcn_wmma_f32_16x16x4_f32(
                    /*neg_a=*/false, a, /*neg_b=*/false, b,
                    /*c_mod=*/(short)0, c, /*reuse_a=*/false, /*reuse_b=*/false);
        }
        #pragma unroll
        for (int r = 0; r < 8; ++r) {
            const int m = (lane < 16) ? r : (r + 8);
            s_atoms[m][ncol + mrow] = c[r] * s_amp[m];
        }
    }
    __syncthreads();

    // ---- stage 3: scatter-add atoms at impulse positions (ds_add_f32 atomics) ----
    for (int t = tid; t < N_ITERS * KERNEL_N; t += 1024) {
        const int i   = t >> 8;
        const int k   = t & 255;
        const int pos = s_pos[i] + k;
        if (pos < N_SAMPLES) atomicAdd(&s_acc[pos], s_atoms[i][k]);
    }
    __syncthreads();

    // ---- stage 4: max-abs reduce + normalize ----
    float mx = 0.0f;
    for (int i = tid; i < N_SAMPLES; i += 1024) mx = fmaxf(mx, fabsf(s_acc[i]));
    #pragma unroll
    for (int off = 16; off > 0; off >>= 1) mx = fmaxf(mx, __shfl_xor(mx, off, 32));
    if (lane == 0) s_red[wave] = mx;
    __syncthreads();
    if (wave == 0) {
        float m = s_red[lane];
        #pragma unroll
        for (int off = 16; off > 0; off >>= 1) m = fmaxf(m, __shfl_xor(m, off, 32));
        if (lane == 0) s_mx = m;
    }
    __syncthreads();

    const float denom = s_mx + 1e-8f;
    for (int i = tid; i < N_SAMPLES; i += 1024) out[i] = s_acc[i] / denom;
}

extern "C" void kernel_launch(void* const* d_in, const int* in_sizes, int n_in,
                              void* d_out, int out_size, void* d_ws, size_t ws_size,
                              hipStream_t stream) {
    (void)in_sizes; (void)n_in; (void)out_size; (void)d_ws; (void)ws_size;
    // setup_inputs order: x (unused), atom_selection, amps, sched_params, d
    const float* atom_selection = (const float*)d_in[1];
    const float* amps           = (const float*)d_in[2];
    const float* sched          = (const float*)d_in[3];
    const float* dict           = (const float*)d_in[4];
    float* out = (float*)d_out;

    hipLaunchKernelGGL(sparse_synth_kernel, dim3(1), dim3(1024), 0, stream,
                       atom_selection, amps, sched, dict, out);
}